// GCN_30391188586951
// MI455X (gfx1250) — compile-verified
//
#include <hip/hip_runtime.h>
#include <hip/hip_bf16.h>

typedef float v2f __attribute__((ext_vector_type(2)));
typedef float v8f __attribute__((ext_vector_type(8)));

#define TILE_M    128
#define KTILE     32
#define XS_STRIDE 36     // 128x32 X tile, padded: bank-conflict-free A-frag reads
#define WP_STRIDE 144    // float2 units per K-pair row (128 + 16 pad): halves hit disjoint banks

// Async global->LDS path (gfx1250), guarded so compile never breaks if absent.
#if defined(__gfx1250__) && __has_builtin(__builtin_amdgcn_global_load_async_to_lds_b128) && \
    __has_builtin(__builtin_amdgcn_global_load_async_to_lds_b32) && \
    __has_builtin(__builtin_amdgcn_s_wait_asynccnt)
#define USE_ASYNC_LDS 1
// Param types per hipcc diagnostics: b128 takes (int4 AS1*, int4 AS3*, Ii, Ii),
// b32 takes (int AS1*, int AS3*, Ii, Ii) -- non-const, typed pointers.
typedef int v4i_ls __attribute__((vector_size(16)));
typedef __attribute__((address_space(1))) v4i_ls* g_v4i_p;
typedef __attribute__((address_space(3))) v4i_ls* l_v4i_p;
typedef __attribute__((address_space(1))) int*    g_i32_p;
typedef __attribute__((address_space(3))) int*    l_i32_p;
#define ASYNC_CPY128(g, l) __builtin_amdgcn_global_load_async_to_lds_b128( \
    (g_v4i_p)(uintptr_t)(g), (l_v4i_p)(uintptr_t)(l), 0, 0)
#define ASYNC_CPY32(g, l)  __builtin_amdgcn_global_load_async_to_lds_b32( \
    (g_i32_p)(uintptr_t)(g), (l_i32_p)(uintptr_t)(l), 0, 0)
#endif

// ---------------------------------------------------------------------------
// WMMA GEMM: out[M x 128] = A[M x 128] * W[128 x 128] (+bias) (+relu)
// 256 threads = 8 waves. Each block does 128 rows; each wave a 16-row strip
// across all 128 output columns (8 accumulators of v8f).
// W staged in LDS pre-paired so each B frag is one ds_load_b64.
// ---------------------------------------------------------------------------
__launch_bounds__(256)
__global__ void gcn_gemm128(const float* __restrict__ A,
                            const float* __restrict__ W,
                            const float* __restrict__ bias,
                            float* __restrict__ out,
                            int M, int relu) {
  __shared__ float Xs[TILE_M * XS_STRIDE];
  __shared__ float Wp[(KTILE / 2) * WP_STRIDE * 2];   // [K/2][n][parity]

  const int tid  = threadIdx.x;
  const int lane = tid & 31;
  const int wave = tid >> 5;
  const int half = lane >> 4;   // 0: lanes 0-15, 1: lanes 16-31
  const int l16  = lane & 15;
  const int rowBase = blockIdx.x * TILE_M;

  v8f acc[8] = {};

  for (int k0 = 0; k0 < 128; k0 += KTILE) {
#ifdef USE_ASYNC_LDS
    // --- X tile: rows [rowBase, rowBase+128), cols [k0,k0+32). 1024 x b128 async.
    for (int i = tid; i < TILE_M * (KTILE / 4); i += 256) {
      int r  = i >> 3;                 // 8 float4 per row
      int c4 = i & 7;
      int gr = rowBase + r;
      if (gr >= M) gr = M - 1;         // clamp: A row r only affects masked-out rows
      ASYNC_CPY128(A + (size_t)gr * 128 + k0 + c4 * 4, &Xs[r * XS_STRIDE + c4 * 4]);
    }
    // --- W tile, scattered into paired layout with per-lane LDS addresses.
    for (int i = tid; i < KTILE * 128; i += 256) {
      int r = i >> 7;                  // K row within tile
      int c = i & 127;                 // N col
      ASYNC_CPY32(W + (size_t)(k0 + r) * 128 + c,
                  &Wp[(r >> 1) * (WP_STRIDE * 2) + c * 2 + (r & 1)]);
    }
    __builtin_amdgcn_s_wait_asynccnt(0);
    __syncthreads();
#else
    for (int i = tid; i < TILE_M * (KTILE / 4); i += 256) {
      int r  = i >> 3;
      int c4 = i & 7;
      float4 v = make_float4(0.f, 0.f, 0.f, 0.f);
      int gr = rowBase + r;
      if (gr < M) v = *(const float4*)(A + (size_t)gr * 128 + k0 + c4 * 4);
      *(float4*)(&Xs[r * XS_STRIDE + c4 * 4]) = v;
    }
    for (int i = tid; i < KTILE * 32; i += 256) {
      int r  = i >> 5;
      int c4 = i & 31;
      float4 v = *(const float4*)(W + (size_t)(k0 + r) * 128 + c4 * 4);
      float* b = &Wp[(r >> 1) * (WP_STRIDE * 2) + (r & 1)];
      b[(c4 * 4 + 0) * 2] = v.x;
      b[(c4 * 4 + 1) * 2] = v.y;
      b[(c4 * 4 + 2) * 2] = v.z;
      b[(c4 * 4 + 3) * 2] = v.w;
    }
    __syncthreads();
#endif

    const int arow = wave * 16 + l16;    // A-matrix M index for this lane
    #pragma unroll
    for (int kk = 0; kk < KTILE; kk += 4) {
      // A 16x4 f32 frag: lane l, vgpr v holds A[l%16][kk + 2*(l/16) + v]
      v2f a = *(const v2f*)(&Xs[arow * XS_STRIDE + kk + 2 * half]);
      // B pair row index: K = kk + 2*half -> pair p = kk/2 + half
      const v2f* wrow = (const v2f*)&Wp[((kk >> 1) + half) * (WP_STRIDE * 2)];
      #pragma unroll
      for (int t = 0; t < 8; ++t) {
        v2f b = wrow[t * 16 + l16];    // single ds_load_b64, feeds wmma directly
        acc[t] = __builtin_amdgcn_wmma_f32_16x16x4_f32(
            /*neg_a=*/false, a, /*neg_b=*/false, b,
            /*c_mod=*/(short)0, acc[t], /*reuse_a=*/false, /*reuse_b=*/false);
      }
    }
    __syncthreads();
  }

  // D layout: vgpr v, lane l -> row (v + 8*(l/16)) within strip, col l%16.
  #pragma unroll
  for (int t = 0; t < 8; ++t) {
    float bval = bias ? bias[t * 16 + l16] : 0.0f;
    #pragma unroll
    for (int v = 0; v < 8; ++v) {
      int grow = rowBase + wave * 16 + v + 8 * half;
      if (grow < M) {
        float val = acc[t][v] + bval;
        if (relu) val = fmaxf(val, 0.0f);
        out[(size_t)grow * 128 + t * 16 + l16] = val;
      }
    }
  }
}

// ---------------------------------------------------------------------------
// Degree / norm kernels
// ---------------------------------------------------------------------------
__global__ void deg_init(float* __restrict__ deg, int n) {
  int i = blockIdx.x * blockDim.x + threadIdx.x;
  if (i < n) deg[i] = 1.0f;  // self loop
}
__global__ void deg_edges(float* __restrict__ deg, const int* __restrict__ dst, int nE) {
  int i = blockIdx.x * blockDim.x + threadIdx.x;
  if (i < nE) atomicAdd(&deg[dst[i]], 1.0f);
}
__global__ void deg_rsqrt(float* __restrict__ deg, int n) {
  int i = blockIdx.x * blockDim.x + threadIdx.x;
  if (i < n) deg[i] = rsqrtf(deg[i]);
}

// ---------------------------------------------------------------------------
// Edge aggregation: one wave32 per edge, each lane a float4 of the 128-f row.
// agg[dst] += H[src] * dinv[src]*dinv[dst]
// ---------------------------------------------------------------------------
__launch_bounds__(256)
__global__ void gcn_edge_agg(const float* __restrict__ H,
                             float* __restrict__ AGG,
                             const int* __restrict__ src,
                             const int* __restrict__ dst,
                             const float* __restrict__ dinv, int nE) {
  int wid  = (int)((blockIdx.x * (unsigned)blockDim.x + threadIdx.x) >> 5);
  int lane = threadIdx.x & 31;
  if (wid >= nE) return;
  int s = src[wid];
  int d = dst[wid];
  float w = dinv[s] * dinv[d];
  float4 hv = *(const float4*)(H + (size_t)s * 128 + lane * 4);
  float* o = AGG + (size_t)d * 128 + lane * 4;
  atomicAdd(o + 0, hv.x * w);
  atomicAdd(o + 1, hv.y * w);
  atomicAdd(o + 2, hv.z * w);
  atomicAdd(o + 3, hv.w * w);
}

// out = [relu](agg + H*dinv^2 + bias)   -- folds the self loop in
__global__ void gcn_finalize(const float* __restrict__ AGG,
                             const float* __restrict__ H,
                             const float* __restrict__ dinv,
                             const float* __restrict__ bias,
                             float* __restrict__ out, int relu, int n) {
  int i = blockIdx.x * blockDim.x + threadIdx.x;
  if (i >= n) return;
  int node = i >> 7;
  int c    = i & 127;
  float dn = dinv[node];
  float v  = AGG[i] + H[i] * dn * dn + bias[c];
  if (relu) v = fmaxf(v, 0.0f);
  out[i] = v;
}

// ---------------------------------------------------------------------------
// Global mean pool
// ---------------------------------------------------------------------------
__launch_bounds__(256)
__global__ void pool_sum(const float* __restrict__ H,
                         const int* __restrict__ batch,
                         float* __restrict__ pooled,
                         float* __restrict__ cnt, int nNodes) {
  int node = (int)((blockIdx.x * (unsigned)blockDim.x + threadIdx.x) >> 5);
  int lane = threadIdx.x & 31;
  if (node >= nNodes) return;
  int g = batch[node];
  float4 hv = *(const float4*)(H + (size_t)node * 128 + lane * 4);
  float* o = pooled + (size_t)g * 128 + lane * 4;
  atomicAdd(o + 0, hv.x);
  atomicAdd(o + 1, hv.y);
  atomicAdd(o + 2, hv.z);
  atomicAdd(o + 3, hv.w);
  if (lane == 0) atomicAdd(&cnt[g], 1.0f);
}

__global__ void pool_mean(float* __restrict__ pooled, const float* __restrict__ cnt, int n) {
  int i = blockIdx.x * blockDim.x + threadIdx.x;
  if (i >= n) return;
  pooled[i] /= fmaxf(cnt[i >> 7], 1.0f);
}

// ---------------------------------------------------------------------------
extern "C" void kernel_launch(void* const* d_in, const int* in_sizes, int n_in,
                              void* d_out, int out_size, void* d_ws, size_t ws_size,
                              hipStream_t stream) {
  const float* x     = (const float*)d_in[0];
  const int*   eidx  = (const int*)d_in[1];
  const int*   batch = (const int*)d_in[2];
  const float* W1 = (const float*)d_in[3];
  const float* b1 = (const float*)d_in[4];
  const float* W2 = (const float*)d_in[5];
  const float* b2 = (const float*)d_in[6];
  const float* W3 = (const float*)d_in[7];
  const float* b3 = (const float*)d_in[8];
  const float* lW = (const float*)d_in[9];
  const float* lb = (const float*)d_in[10];
  float* out = (float*)d_out;

  const int NN = in_sizes[0] / 128;   // 100000
  const int NE = in_sizes[1] / 2;     // 1600000
  const int G  = out_size / 128;      // 512

  const int* src = eidx;        // edge_index[0]
  const int* dst = eidx + NE;   // edge_index[1]

  // workspace layout (floats), all sub-offsets keep 16B alignment
  float* bufA   = (float*)d_ws;                     // NN*128
  float* bufB   = bufA + (size_t)NN * 128;          // NN*128
  float* dinv   = bufB + (size_t)NN * 128;          // NN
  float* pooled = dinv + (size_t)NN;                // G*128
  float* cnt    = pooled + (size_t)G * 128;         // G

  const size_t featBytes = (size_t)NN * 128 * sizeof(float);
  const int nElem   = NN * 128;
  const dim3 blk256(256);
  const int gemmBlocks = (NN + TILE_M - 1) / TILE_M;
  const int edgeBlocks = (NE + 7) / 8;   // 8 waves per block, 1 edge per wave
  const int nodeBlocks = (NN + 7) / 8;
  const int elemBlocks = (nElem + 255) / 256;

  // --- degree normalization ---
  deg_init<<<(NN + 255) / 256, blk256, 0, stream>>>(dinv, NN);
  deg_edges<<<(NE + 255) / 256, blk256, 0, stream>>>(dinv, dst, NE);
  deg_rsqrt<<<(NN + 255) / 256, blk256, 0, stream>>>(dinv, NN);

  // --- layer 1: H0 = x@W1 ; agg ; relu(+b1) ---
  gcn_gemm128<<<gemmBlocks, blk256, 0, stream>>>(x, W1, nullptr, bufA, NN, 0);
  (void)hipMemsetAsync(bufB, 0, featBytes, stream);
  gcn_edge_agg<<<edgeBlocks, blk256, 0, stream>>>(bufA, bufB, src, dst, dinv, NE);
  gcn_finalize<<<elemBlocks, blk256, 0, stream>>>(bufB, bufA, dinv, b1, bufA, 1, nElem);

  // --- layer 2 ---
  gcn_gemm128<<<gemmBlocks, blk256, 0, stream>>>(bufA, W2, nullptr, bufB, NN, 0);
  (void)hipMemsetAsync(bufA, 0, featBytes, stream);
  gcn_edge_agg<<<edgeBlocks, blk256, 0, stream>>>(bufB, bufA, src, dst, dinv, NE);
  gcn_finalize<<<elemBlocks, blk256, 0, stream>>>(bufA, bufB, dinv, b2, bufB, 1, nElem);

  // --- layer 3 (no relu) ---
  gcn_gemm128<<<gemmBlocks, blk256, 0, stream>>>(bufB, W3, nullptr, bufA, NN, 0);
  (void)hipMemsetAsync(bufB, 0, featBytes, stream);
  gcn_edge_agg<<<edgeBlocks, blk256, 0, stream>>>(bufA, bufB, src, dst, dinv, NE);
  gcn_finalize<<<elemBlocks, blk256, 0, stream>>>(bufB, bufA, dinv, b3, bufB, 0, nElem);

  // --- global mean pool ---
  (void)hipMemsetAsync(pooled, 0, (size_t)G * 128 * sizeof(float), stream);
  (void)hipMemsetAsync(cnt, 0, (size_t)G * sizeof(float), stream);
  pool_sum<<<nodeBlocks, blk256, 0, stream>>>(bufB, batch, pooled, cnt, NN);
  pool_mean<<<(G * 128 + 255) / 256, blk256, 0, stream>>>(pooled, cnt, G * 128);

  // --- final linear: out = pooled @ lin_W + lin_b ---
  gcn_gemm128<<<(G + TILE_M - 1) / TILE_M, blk256, 0, stream>>>(pooled, lW, lb, out, G, 0);
}